// AttentionMM_48687749267786
// MI455X (gfx1250) — compile-verified
//
#include <hip/hip_runtime.h>
#include <math.h>

#define B_ 32
#define T_ 2048
#define E_ 128

typedef __attribute__((ext_vector_type(2))) float v2f;
typedef __attribute__((ext_vector_type(8))) float v8f;

// ---------------------------------------------------------------------------
// 0) zero-init scratch accumulators and output (ws/out are poisoned by harness)
// ---------------------------------------------------------------------------
__global__ void init_zero_kernel(float* __restrict__ s1, float* __restrict__ s2,
                                 float* __restrict__ c1, float* __restrict__ c2,
                                 float* __restrict__ out) {
    int i = blockIdx.x * 256 + threadIdx.x;          // 8192 threads
    if (i < B_ * E_) { s1[i] = 0.f; s2[i] = 0.f; c1[i] = 0.f; c2[i] = 0.f; }
    if (i < B_ * 2 * E_) out[i] = 0.f;
}

// ---------------------------------------------------------------------------
// 1) s1[b,e] = sum_t x1[b,t,e]; s2[b,e] = sum_t x2[b,t,e]
//    grid (16 chunks, 32 batches), block 256: tid<128 -> x1, else x2
// ---------------------------------------------------------------------------
__global__ void colsum_kernel(const float* __restrict__ x1, const float* __restrict__ x2,
                              float* __restrict__ s1, float* __restrict__ s2) {
    int b = blockIdx.y, chunk = blockIdx.x, tid = threadIdx.x;
    const float* x = (tid < E_) ? x1 : x2;
    float* s = (tid < E_) ? s1 : s2;
    int e = tid & (E_ - 1);
    const float* base = x + ((size_t)b * T_ + (size_t)chunk * 128) * E_ + e;
    float acc = 0.f;
#pragma unroll 8
    for (int r = 0; r < 128; ++r) acc += base[(size_t)r * E_];
    atomicAdd(&s[b * E_ + e], acc);
}

// ---------------------------------------------------------------------------
// 2) fused pass over x1/x2 rows:
//    t1 = x1row.s2 ; t2 = x2row.s1
//    c1 += t1*x2row/T ; c2 += t2*x1row/T
//    xw1[b,t] = x1row.W1 + b1[t] ; xw2[b,t] = x2row.W2 + b2[t]
//    grid (16 chunks, 32 batches), block 256 = 8 waves, 16 rows/wave
// ---------------------------------------------------------------------------
__global__ void center_kernel(const float* __restrict__ x1, const float* __restrict__ x2,
                              const float* __restrict__ s1, const float* __restrict__ s2,
                              const float* __restrict__ W1, const float* __restrict__ W2,
                              const float* __restrict__ b1, const float* __restrict__ b2,
                              float* __restrict__ c1, float* __restrict__ c2,
                              float* __restrict__ xw1, float* __restrict__ xw2) {
    int b = blockIdx.y, chunk = blockIdx.x;
    int tid = threadIdx.x, lane = tid & 31, wave = tid >> 5;
    int e4 = lane * 4;                                 // each lane owns 4 e's

    __shared__ float c1s[E_], c2s[E_];
    if (tid < E_) { c1s[tid] = 0.f; c2s[tid] = 0.f; }
    __syncthreads();

    const float4 s1r = *(const float4*)(s1 + b * E_ + e4);
    const float4 s2r = *(const float4*)(s2 + b * E_ + e4);
    const float4 w1r = *(const float4*)(W1 + e4);
    const float4 w2r = *(const float4*)(W2 + e4);

    float4 acc1 = make_float4(0.f, 0.f, 0.f, 0.f);
    float4 acc2 = make_float4(0.f, 0.f, 0.f, 0.f);

    int t0 = chunk * 128 + wave * 16;
    for (int r = 0; r < 16; ++r) {
        int t = t0 + r;
        size_t row = ((size_t)b * T_ + t) * E_;
        const float4 x1v = *(const float4*)(x1 + row + e4);
        const float4 x2v = *(const float4*)(x2 + row + e4);
        float p1  = x1v.x * s2r.x + x1v.y * s2r.y + x1v.z * s2r.z + x1v.w * s2r.w;
        float p2  = x2v.x * s1r.x + x2v.y * s1r.y + x2v.z * s1r.z + x2v.w * s1r.w;
        float pw1 = x1v.x * w1r.x + x1v.y * w1r.y + x1v.z * w1r.z + x1v.w * w1r.w;
        float pw2 = x2v.x * w2r.x + x2v.y * w2r.y + x2v.z * w2r.z + x2v.w * w2r.w;
        // wave32 butterfly reduction -> every lane holds full dot products
        for (int off = 16; off > 0; off >>= 1) {
            p1  += __shfl_xor(p1,  off, 32);
            p2  += __shfl_xor(p2,  off, 32);
            pw1 += __shfl_xor(pw1, off, 32);
            pw2 += __shfl_xor(pw2, off, 32);
        }
        if (lane == 0) {
            xw1[(size_t)b * T_ + t] = pw1 + b1[t];
            xw2[(size_t)b * T_ + t] = pw2 + b2[t];
        }
        acc1.x += p1 * x2v.x; acc1.y += p1 * x2v.y; acc1.z += p1 * x2v.z; acc1.w += p1 * x2v.w;
        acc2.x += p2 * x1v.x; acc2.y += p2 * x1v.y; acc2.z += p2 * x1v.z; acc2.w += p2 * x1v.w;
    }
    atomicAdd(&c1s[e4 + 0], acc1.x); atomicAdd(&c1s[e4 + 1], acc1.y);
    atomicAdd(&c1s[e4 + 2], acc1.z); atomicAdd(&c1s[e4 + 3], acc1.w);
    atomicAdd(&c2s[e4 + 0], acc2.x); atomicAdd(&c2s[e4 + 1], acc2.y);
    atomicAdd(&c2s[e4 + 2], acc2.z); atomicAdd(&c2s[e4 + 3], acc2.w);
    __syncthreads();
    const float invT = 1.0f / (float)T_;
    if (tid < E_) {
        atomicAdd(&c1[b * E_ + tid], c1s[tid] * invT);
        atomicAdd(&c2[b * E_ + tid], c2s[tid] * invT);
    }
}

// ---------------------------------------------------------------------------
// 3) et = c @ U + xw (+ bias), via V_WMMA_F32_16X16X4_F32.
//    One wave per 16x16 output tile; K=128 -> 32 WMMA ops.
//    grid (128 n-tiles, 2 m-tiles, 2 sides), block 32 (EXEC all ones).
//    In-place: et buffer aliases xw buffer (read-before-write per element).
// ---------------------------------------------------------------------------
__global__ void et_wmma_kernel(const float* __restrict__ c1, const float* __restrict__ c2,
                               const float* __restrict__ U1, const float* __restrict__ U2,
                               const float* __restrict__ b1, const float* __restrict__ b2,
                               float* __restrict__ et1, float* __restrict__ et2) {
    const int which = blockIdx.z;
    const float* C    = which ? c2  : c1;
    const float* U    = which ? U2  : U1;
    const float* bias = which ? b2  : b1;
    float* et         = which ? et2 : et1;

    const int lane  = threadIdx.x;     // 0..31
    const int half  = lane >> 4;       // 0 / 1
    const int l16   = lane & 15;
    const int mbase = blockIdx.y * 16;
    const int nbase = blockIdx.x * 16;

    const float* Crow = C + (mbase + l16) * E_;        // A row for this lane
    v8f acc = {};                                       // C starts at 0
#pragma unroll 8
    for (int k = 0; k < E_; k += 4) {
        // A 16x4 f32 layout: VGPR0 = K=(0|2), VGPR1 = K=(1|3), half-split on lanes
        v2f a;
        a.x = Crow[k + 2 * half];
        a.y = Crow[k + 2 * half + 1];
        // B 4x16 f32 layout (mirror): lanes index N, half-split on K
        v2f bv;
        bv.x = U[(size_t)(k + 2 * half)     * T_ + nbase + l16];
        bv.y = U[(size_t)(k + 2 * half + 1) * T_ + nbase + l16];
        acc = __builtin_amdgcn_wmma_f32_16x16x4_f32(
            /*neg_a=*/false, a, /*neg_b=*/false, bv,
            /*c_mod=*/(short)0, acc, /*reuse_a=*/false, /*reuse_b=*/false);
    }
    // D layout: VGPR v -> row (v + 8*half), N = l16
#pragma unroll
    for (int v = 0; v < 8; ++v) {
        int mm = mbase + v + 8 * half;
        int nn = nbase + l16;
        size_t idx = (size_t)mm * T_ + nn;
        et[idx] = acc[v] + et[idx] + bias[nn];          // et aliases xw: read-then-write
    }
}

// ---------------------------------------------------------------------------
// 4) row softmax over T, in place.  grid (32 batches, 2 sides), block 256.
// ---------------------------------------------------------------------------
__global__ void softmax_kernel(float* __restrict__ et1, float* __restrict__ et2) {
    float* row = (blockIdx.y ? et2 : et1) + (size_t)blockIdx.x * T_;
    int tid = threadIdx.x;
    __shared__ float red[256];
    float m = -INFINITY;
    for (int i = tid; i < T_; i += 256) m = fmaxf(m, row[i]);
    red[tid] = m; __syncthreads();
    for (int s = 128; s > 0; s >>= 1) { if (tid < s) red[tid] = fmaxf(red[tid], red[tid + s]); __syncthreads(); }
    m = red[0]; __syncthreads();
    float sum = 0.f;
    for (int i = tid; i < T_; i += 256) sum += __expf(row[i] - m);
    red[tid] = sum; __syncthreads();
    for (int s = 128; s > 0; s >>= 1) { if (tid < s) red[tid] += red[tid + s]; __syncthreads(); }
    float inv = 1.0f / red[0];
    for (int i = tid; i < T_; i += 256) row[i] = __expf(row[i] - m) * inv;
}

// ---------------------------------------------------------------------------
// 5) o1[b,:] = sum_t at1[b,t]*x1[b,t,:]; o2 likewise; out = concat(o1,o2)
//    grid (16 chunks, 32 batches), block 256 = 8 waves, 16 rows/wave
// ---------------------------------------------------------------------------
__global__ void wsum_kernel(const float* __restrict__ x1, const float* __restrict__ x2,
                            const float* __restrict__ at1, const float* __restrict__ at2,
                            float* __restrict__ out) {
    int b = blockIdx.y, chunk = blockIdx.x;
    int tid = threadIdx.x, lane = tid & 31, wave = tid >> 5;
    int e4 = lane * 4;
    __shared__ float o1s[E_], o2s[E_];
    if (tid < E_) { o1s[tid] = 0.f; o2s[tid] = 0.f; }
    __syncthreads();

    float4 acc1 = make_float4(0.f, 0.f, 0.f, 0.f);
    float4 acc2 = make_float4(0.f, 0.f, 0.f, 0.f);
    int t0 = chunk * 128 + wave * 16;
    for (int r = 0; r < 16; ++r) {
        int t = t0 + r;
        float a1 = at1[(size_t)b * T_ + t];
        float a2 = at2[(size_t)b * T_ + t];
        size_t row = ((size_t)b * T_ + t) * E_;
        const float4 x1v = *(const float4*)(x1 + row + e4);
        const float4 x2v = *(const float4*)(x2 + row + e4);
        acc1.x += a1 * x1v.x; acc1.y += a1 * x1v.y; acc1.z += a1 * x1v.z; acc1.w += a1 * x1v.w;
        acc2.x += a2 * x2v.x; acc2.y += a2 * x2v.y; acc2.z += a2 * x2v.z; acc2.w += a2 * x2v.w;
    }
    atomicAdd(&o1s[e4 + 0], acc1.x); atomicAdd(&o1s[e4 + 1], acc1.y);
    atomicAdd(&o1s[e4 + 2], acc1.z); atomicAdd(&o1s[e4 + 3], acc1.w);
    atomicAdd(&o2s[e4 + 0], acc2.x); atomicAdd(&o2s[e4 + 1], acc2.y);
    atomicAdd(&o2s[e4 + 2], acc2.z); atomicAdd(&o2s[e4 + 3], acc2.w);
    __syncthreads();
    if (tid < E_) {
        atomicAdd(&out[(size_t)b * 2 * E_ + tid],        o1s[tid]);
        atomicAdd(&out[(size_t)b * 2 * E_ + E_ + tid],   o2s[tid]);
    }
}

// ---------------------------------------------------------------------------
extern "C" void kernel_launch(void* const* d_in, const int* in_sizes, int n_in,
                              void* d_out, int out_size, void* d_ws, size_t ws_size,
                              hipStream_t stream) {
    const float* x1 = (const float*)d_in[0];
    const float* x2 = (const float*)d_in[1];
    const float* W1 = (const float*)d_in[2];
    const float* b1 = (const float*)d_in[3];
    const float* U1 = (const float*)d_in[4];
    const float* W2 = (const float*)d_in[5];
    const float* b2 = (const float*)d_in[6];
    const float* U2 = (const float*)d_in[7];
    float* out = (float*)d_out;

    float* ws  = (float*)d_ws;
    float* s1  = ws;                    // B*E
    float* s2  = ws + 4096;             // B*E
    float* c1  = ws + 8192;             // B*E
    float* c2  = ws + 12288;            // B*E
    float* xw1 = ws + 16384;            // B*T (then et1, then at1, in place)
    float* xw2 = ws + 16384 + B_ * T_;  // B*T (then et2, then at2, in place)

    init_zero_kernel<<<32, 256, 0, stream>>>(s1, s2, c1, c2, out);
    colsum_kernel<<<dim3(16, B_), 256, 0, stream>>>(x1, x2, s1, s2);
    center_kernel<<<dim3(16, B_), 256, 0, stream>>>(x1, x2, s1, s2, W1, W2, b1, b2,
                                                    c1, c2, xw1, xw2);
    et_wmma_kernel<<<dim3(T_ / 16, 2, 2), 32, 0, stream>>>(c1, c2, U1, U2, b1, b2, xw1, xw2);
    softmax_kernel<<<dim3(B_, 2), 256, 0, stream>>>(xw1, xw2);
    wsum_kernel<<<dim3(16, B_), 256, 0, stream>>>(x1, x2, xw1, xw2, out);
}